// ConfusionMatrixMetric_28484223107942
// MI455X (gfx1250) — compile-verified
//
#include <hip/hip_runtime.h>
#include <cfloat>

// Native vector type so __builtin_nontemporal_load lowers to
// global_load_b128 th:TH_LOAD_NT on gfx1250 (stream-once data, bypass L2 retention).
typedef float v4f __attribute__((ext_vector_type(4)));

__global__ __launch_bounds__(256) void cm_zero_kernel(float* __restrict__ out, int n) {
    int i  = blockIdx.x * blockDim.x + threadIdx.x;
    int i4 = i << 2;
    if (i4 + 3 < n) {
        v4f z = {0.f, 0.f, 0.f, 0.f};
        *(v4f*)(out + i4) = z;
    } else {
        for (int k = i4; k < n; ++k) out[k] = 0.f;
    }
}

// One wave32 per row. Lane i consumes float4 chunks i, i+32, ... (coalesced
// 512B/wave requests), keeps a running (max, argmax), then a 5-step xor
// butterfly reduces across the wave. Lane 0 scatter-adds into the histogram.
__global__ __launch_bounds__(256) void cm_argmax_hist_kernel(
    const float* __restrict__ pred,
    const int*   __restrict__ target,
    float*       __restrict__ out,
    int nrows, int C) {
    const int lane = threadIdx.x & 31;
    const int wave = threadIdx.x >> 5;
    const int row  = blockIdx.x * (blockDim.x >> 5) + wave;
    if (row >= nrows) return;

    const v4f* rp = (const v4f*)(pred + (size_t)row * (size_t)C);
    const int  nq = C >> 2;   // float4 chunks per row (C % 4 == 0 path)

    float bestv = -FLT_MAX;
    int   besti = 0;

#define CM_STEP(vec, qq)                                              \
    {                                                                 \
        int c_ = (qq) << 2;                                           \
        if ((vec)[0] > bestv) { bestv = (vec)[0]; besti = c_;     }   \
        if ((vec)[1] > bestv) { bestv = (vec)[1]; besti = c_ + 1; }   \
        if ((vec)[2] > bestv) { bestv = (vec)[2]; besti = c_ + 2; }   \
        if ((vec)[3] > bestv) { bestv = (vec)[3]; besti = c_ + 3; }   \
    }

    int q = lane;
    // 4x unroll: keep >=4 outstanding B128 NT loads per lane for latency hiding.
    for (; q + 96 < nq; q += 128) {
        v4f a0 = __builtin_nontemporal_load(rp + q);
        v4f a1 = __builtin_nontemporal_load(rp + q + 32);
        v4f a2 = __builtin_nontemporal_load(rp + q + 64);
        v4f a3 = __builtin_nontemporal_load(rp + q + 96);
        CM_STEP(a0, q)
        CM_STEP(a1, q + 32)
        CM_STEP(a2, q + 64)
        CM_STEP(a3, q + 96)
    }
    for (; q < nq; q += 32) {
        v4f a = __builtin_nontemporal_load(rp + q);
        CM_STEP(a, q)
    }
#undef CM_STEP

    // Wave32 butterfly argmax; lowest index wins ties (matches jnp.argmax).
#pragma unroll
    for (int off = 16; off > 0; off >>= 1) {
        float ov = __shfl_xor(bestv, off, 32);
        int   oi = __shfl_xor(besti, off, 32);
        if (ov > bestv || (ov == bestv && oi < besti)) { bestv = ov; besti = oi; }
    }

    if (lane == 0) {
        int t = target[row];
        atomicAdd(out + (size_t)t * (size_t)C + besti, 1.0f);
    }
}

// Scalar fallback for C not divisible by 4 (unused at C=1000, kept for safety).
__global__ __launch_bounds__(256) void cm_argmax_hist_scalar_kernel(
    const float* __restrict__ pred,
    const int*   __restrict__ target,
    float*       __restrict__ out,
    int nrows, int C) {
    const int lane = threadIdx.x & 31;
    const int wave = threadIdx.x >> 5;
    const int row  = blockIdx.x * (blockDim.x >> 5) + wave;
    if (row >= nrows) return;

    const float* rp = pred + (size_t)row * (size_t)C;
    float bestv = -FLT_MAX;
    int   besti = 0;
    for (int c = lane; c < C; c += 32) {
        float v = __builtin_nontemporal_load(rp + c);
        if (v > bestv) { bestv = v; besti = c; }
    }
#pragma unroll
    for (int off = 16; off > 0; off >>= 1) {
        float ov = __shfl_xor(bestv, off, 32);
        int   oi = __shfl_xor(besti, off, 32);
        if (ov > bestv || (ov == bestv && oi < besti)) { bestv = ov; besti = oi; }
    }
    if (lane == 0) {
        int t = target[row];
        atomicAdd(out + (size_t)t * (size_t)C + besti, 1.0f);
    }
}

extern "C" void kernel_launch(void* const* d_in, const int* in_sizes, int n_in,
                              void* d_out, int out_size, void* d_ws, size_t ws_size,
                              hipStream_t stream) {
    (void)n_in; (void)d_ws; (void)ws_size;

    const float* pred   = (const float*)d_in[0];
    const int*   target = (const int*)d_in[1];
    float*       out    = (float*)d_out;

    const int nrows = in_sizes[1];                 // N = 262144
    const int C     = in_sizes[0] / in_sizes[1];   // C = 1000

    // 1) Zero the (poisoned) histogram output.
    {
        int n4     = (out_size + 3) >> 2;
        int blocks = (n4 + 255) / 256;
        cm_zero_kernel<<<blocks, 256, 0, stream>>>(out, out_size);
    }

    // 2) Argmax + scatter-add. 8 rows (waves) per 256-thread block.
    {
        int rows_per_block = 256 / 32;
        int blocks = (nrows + rows_per_block - 1) / rows_per_block;
        if ((C & 3) == 0) {
            cm_argmax_hist_kernel<<<blocks, 256, 0, stream>>>(pred, target, out, nrows, C);
        } else {
            cm_argmax_hist_scalar_kernel<<<blocks, 256, 0, stream>>>(pred, target, out, nrows, C);
        }
    }
}